// GlobalPolarHead_36318243455486
// MI455X (gfx1250) — compile-verified
//
#include <hip/hip_runtime.h>
#include <math.h>

typedef __attribute__((ext_vector_type(2))) float v2f;
typedef __attribute__((ext_vector_type(4))) float v4f;

#define IMG_W_F 800.0f
#define IMG_H_F 320.0f
#define NOFF 72
#define NFEAT 36

// Shared per-(anchor, k-pair) tail: one thread covers output offsets k0=2t, k0+1.
// Reference axis reversal folded into index math (j = 71 - k). All arithmetic uses
// explicit *_rn intrinsics so rounding matches the reference's mul -> sub -> div
// sequence (no FMA contraction, true division).
__device__ __forceinline__ void polar_tail(
    int a, int t, float s, float c, float r,
    const float* __restrict__ off,
    float* __restrict__ out_feat,
    float* __restrict__ out_anchor,
    float* __restrict__ out_lane)
{
  int k0 = 2 * t;            // output offset indices k0, k0+1
  int j0 = 71 - k0;          // source ray indices j0, j0-1

  const float STEP = 1.0f / 71.0f;   // linspace(0,1,72) step
  float y0 = __fmul_rn(__fmul_rn(STEP, (float)j0),       IMG_H_F);
  float y1 = __fmul_rn(__fmul_rn(STEP, (float)(j0 - 1)), IMG_H_F);
  float x0 = __fdiv_rn(__fsub_rn(r, __fmul_rn(y0, s)), c);
  float x1 = __fdiv_rn(__fsub_rn(r, __fmul_rn(y1, s)), c);

  int base = a * (NOFF * 2) + 4 * t; // float index into (A,72,2), 16B aligned

  // anchor_samples[a, k, :] = (x_{71-k}, ys_{71-k})  -- streaming NT b128
  v4f av = { x0, y0, x1, y1 };
  __builtin_nontemporal_store(av, reinterpret_cast<v4f*>(out_anchor + base));

  // read-once offset stream, NT b64
  v2f offs = __builtin_nontemporal_load(reinterpret_cast<const v2f*>(off) + (a * 36 + t));
  float ly0 = __fsub_rn(IMG_H_F, y0);
  float ly1 = __fsub_rn(IMG_H_F, y1);
  v4f lv = { __fadd_rn(x0, __fmul_rn(offs.x, IMG_W_F)), ly0,
             __fadd_rn(x1, __fmul_rn(offs.y, IMG_W_F)), ly1 };
  __builtin_nontemporal_store(lv, reinterpret_cast<v4f*>(out_lane + base));

  // feat gather: inverse of the strictly-increasing 36-of-72 index set.
  // Thread covering output k writes slot m = 35 - i iff i = ceil(35k/71)
  // satisfies floor(71i/35) == k (interval width 35/71 < 1 => unique).
#pragma unroll
  for (int u = 0; u < 2; ++u) {
    int k = k0 + u;
    int i = (35 * k + 70) / 71;
    if ((71 * i) / 35 == k) {
      int m = 35 - i;
      float x  = u ? x1  : x0;
      float ly = u ? ly1 : ly0;
      v2f fv = { __fdiv_rn(x, IMG_W_F), __fdiv_rn(ly, IMG_H_F) };
      __builtin_nontemporal_store(fv, reinterpret_cast<v2f*>(out_feat) + (a * 36 + m));
    }
  }
}

// Pass 1: one thread per anchor; all transcendental work done exactly once.
__global__ __launch_bounds__(256) void polar_trig_kernel(
    const float* __restrict__ emb, v4f* __restrict__ ws, int num_anchors)
{
  int a = blockIdx.x * 256 + threadIdx.x;
  if (a >= num_anchors) return;
  v2f e = reinterpret_cast<const v2f*>(emb)[a];
  float theta = __fmul_rn(e.x, 3.14159274101257324f);  // float(pi)
  v4f w = { sinf(theta), cosf(theta), __fmul_rn(e.y, IMG_W_F), 0.0f };
  ws[a] = w;
}

// Pass 2: pure streaming; per-anchor {s,c,r} record is a cached broadcast b128.
__global__ __launch_bounds__(256) void polar_stream_kernel(
    const v4f* __restrict__ ws,
    const float* __restrict__ off,
    float* __restrict__ out_feat,
    float* __restrict__ out_anchor,
    float* __restrict__ out_lane,
    int total)
{
  int n = blockIdx.x * 256 + threadIdx.x;
  if (n >= total) return;
  int a = n / 36;
  int t = n - a * 36;
  v4f w = ws[a];
  polar_tail(a, t, w.x, w.y, w.z, off, out_feat, out_anchor, out_lane);
}

// Fallback: fused single pass (used only if d_ws is too small).
__global__ __launch_bounds__(256) void polar_fused_kernel(
    const float* __restrict__ emb,
    const float* __restrict__ off,
    float* __restrict__ out_feat,
    float* __restrict__ out_anchor,
    float* __restrict__ out_lane,
    int total)
{
  int n = blockIdx.x * 256 + threadIdx.x;
  if (n >= total) return;
  int a = n / 36;
  int t = n - a * 36;
  v2f e = reinterpret_cast<const v2f*>(emb)[a];
  float theta = __fmul_rn(e.x, 3.14159274101257324f);
  polar_tail(a, t, sinf(theta), cosf(theta), __fmul_rn(e.y, IMG_W_F),
             off, out_feat, out_anchor, out_lane);
}

extern "C" void kernel_launch(void* const* d_in, const int* in_sizes, int n_in,
                              void* d_out, int out_size, void* d_ws, size_t ws_size,
                              hipStream_t stream) {
  (void)n_in; (void)out_size;

  const float* emb = (const float*)d_in[0];   // (32,4000,2)
  const float* off = (const float*)d_in[1];   // (32,4000,72)

  int num_anchors  = in_sizes[0] / 2;          // 128000
  int feat_elems   = num_anchors * NFEAT * 2;  // 9,216,000
  int anchor_elems = num_anchors * NOFF * 2;   // 18,432,000

  float* out        = (float*)d_out;
  float* out_feat   = out;                     // return order: feat, anchor, lane
  float* out_anchor = out + feat_elems;
  float* out_lane   = out + feat_elems + anchor_elems;

  int total  = num_anchors * 36;               // 4,608,000 threads
  int blocks = (total + 255) / 256;

  if (ws_size >= (size_t)num_anchors * sizeof(v4f)) {
    v4f* ws = (v4f*)d_ws;
    int tb = (num_anchors + 255) / 256;
    polar_trig_kernel<<<tb, 256, 0, stream>>>(emb, ws, num_anchors);
    polar_stream_kernel<<<blocks, 256, 0, stream>>>(ws, off, out_feat, out_anchor,
                                                    out_lane, total);
  } else {
    polar_fused_kernel<<<blocks, 256, 0, stream>>>(emb, off, out_feat, out_anchor,
                                                   out_lane, total);
  }
}